// Embed2GraphByLinear_1202590843558
// MI455X (gfx1250) — compile-verified
//
#include <hip/hip_runtime.h>

typedef float v2f   __attribute__((ext_vector_type(2)));
typedef float v4f   __attribute__((ext_vector_type(4)));
typedef float v8f   __attribute__((ext_vector_type(8)));
typedef float f32x2 __attribute__((ext_vector_type(2)));

#define BB 16
#define RR 360
#define DD 64
#define NN (BB * RR)          // 5760 rows total

// ---- VOP3P packed-f32 helpers (no builtins for these; inline asm) ----------
__device__ __forceinline__ f32x2 pk_add(f32x2 a, f32x2 b) {
    f32x2 d;
    asm("v_pk_add_f32 %0, %1, %2" : "=v"(d) : "v"(a), "v"(b));
    return d;
}
__device__ __forceinline__ void pk_fma(f32x2& acc, f32x2 a, f32x2 b) {
    asm("v_pk_fma_f32 %0, %1, %2, %0" : "+v"(acc) : "v"(a), "v"(b));
}

// Zero-copy split of a 128-bit vector register into two even-aligned pairs.
union V4u { v4f v; f32x2 h[2]; };

// ---------------------------------------------------------------------------
// Stage 1: S[n,d]  = sum_k X[n,k] * W_out[k,d]
//          Rv[n,d] = sum_k X[n,k] * W_out[64+k,d] + b_out[d]
// One wave handles 16 rows, full 64-wide N for both S and Rv (8 WMMA tiles),
// K covered by 16 steps of V_WMMA_F32_16X16X4_F32.  (wmma=128 confirmed.)
// ---------------------------------------------------------------------------
__global__ __launch_bounds__(128) void stage1_gemm_wmma(
    const float* __restrict__ X,      // (5760, 64) row-major
    const float* __restrict__ Wout,   // (128, 64)  row-major
    const float* __restrict__ bout,   // (64)
    float* __restrict__ S,            // (5760, 64)
    float* __restrict__ Rv)           // (5760, 64)
{
    const int wave = threadIdx.x >> 5;
    const int lane = threadIdx.x & 31;
    const int m    = lane & 15;
    const int kh   = lane >> 4;                     // 0 or 1
    const int n0   = (blockIdx.x * 4 + wave) * 16;  // 16-row strip base

    const v8f vzero = {0.f,0.f,0.f,0.f,0.f,0.f,0.f,0.f};
    v8f accS[4], accR[4];
#pragma unroll
    for (int t = 0; t < 4; ++t) { accS[t] = vzero; accR[t] = vzero; }

    const float* xrow = X + (size_t)(n0 + m) * DD;

#pragma unroll
    for (int k = 0; k < 16; ++k) {
        const int kb = 4 * k + 2 * kh;              // K index of this lane's pair
        const v2f a = *(const v2f*)(xrow + kb);     // X[m][kb], X[m][kb+1]

#pragma unroll
        for (int t = 0; t < 4; ++t) {
            const int ncol = t * 16 + m;
            v2f bs, br;
            bs.x = Wout[(size_t)(kb     ) * DD + ncol];
            bs.y = Wout[(size_t)(kb +  1) * DD + ncol];
            br.x = Wout[(size_t)(kb + 64) * DD + ncol];
            br.y = Wout[(size_t)(kb + 65) * DD + ncol];
            accS[t] = __builtin_amdgcn_wmma_f32_16x16x4_f32(
                false, a, false, bs, (short)0, accS[t], false, false);
            accR[t] = __builtin_amdgcn_wmma_f32_16x16x4_f32(
                false, a, false, br, (short)0, accR[t], false, false);
        }
    }

    // Write back; fold b_out into Rv so stage 2 only does add+relu+fma.
#pragma unroll
    for (int t = 0; t < 4; ++t) {
        const int ncol  = t * 16 + m;
        const float bia = bout[ncol];
#pragma unroll
        for (int v = 0; v < 8; ++v) {
            const int row = n0 + v + 8 * kh;
            S [(size_t)row * DD + ncol] = accS[t][v];
            Rv[(size_t)row * DD + ncol] = accR[t][v] + bia;
        }
    }
}

// ---------------------------------------------------------------------------
// Stage 2: out[b,i,j] = relu( sum_d relu(S[b,j,d] + Rv[b,i,d]) * W_cat[d] + b_cat )
// One wave <-> one receiver row i: Rv[b,i,:] and W_cat live in register pairs
// for the whole kernel (no per-q LDS reads for them, no integer division).
// S staged through LDS in 120-row chunks (padded stride 68 floats). Inner
// step per 4 elements: 1 ds_load_b128 + 2 v_pk_add_f32 + 4 v_max + 2
// v_pk_fma_f32. Next S chunk prefetched (global_prefetch_b8) during compute.
// ---------------------------------------------------------------------------
#define JC   120
#define PADW 68
#define IB   8

__global__ __launch_bounds__(256) void stage2_pair(
    const float* __restrict__ S,
    const float* __restrict__ Rv,
    const float* __restrict__ Wcat,   // (64)
    const float* __restrict__ bcat,   // (1)
    float* __restrict__ out)          // (16, 360, 360)
{
    __shared__ float sS[JC * PADW];   // ~31.9 KB

    const int b     = blockIdx.y;
    const int ibase = blockIdx.x * IB;
    const int tid   = threadIdx.x;
    const int wave  = tid >> 5;       // 0..7 -> receiver row  i = ibase + wave
    const int lane  = tid & 31;

    // W_cat -> 32 register pairs (uniform broadcast loads, zero-copy split)
    f32x2 w2[32];
#pragma unroll
    for (int q = 0; q < 16; ++q) {
        V4u u; u.v = ((const v4f*)Wcat)[q];
        w2[2 * q]     = u.h[0];
        w2[2 * q + 1] = u.h[1];
    }
    const float bc = bcat[0];

    // Rv row for this wave -> 32 register pairs (wave-uniform address)
    const float* rrow = Rv + (size_t)(b * RR + ibase + wave) * DD;
    f32x2 r2[32];
#pragma unroll
    for (int q = 0; q < 16; ++q) {
        V4u u; u.v = ((const v4f*)rrow)[q];
        r2[2 * q]     = u.h[0];
        r2[2 * q + 1] = u.h[1];
    }

    float* orow = out + (size_t)(b * RR + ibase + wave) * RR;

    for (int c = 0; c < 3; ++c) {
        const int jbase = c * JC;

        __syncthreads();   // protect sS reuse across chunks
        // S chunk (120 rows x 64) -> LDS, 256 threads cooperative
        for (int idx = tid; idx < JC * 16; idx += 256) {
            const int row = idx >> 4, c4 = idx & 15;
            const v4f v =
                ((const v4f*)(S + (size_t)(b * RR + jbase + row) * DD))[c4];
            *((v4f*)(sS + row * PADW + c4 * 4)) = v;
        }
        __syncthreads();

        // Prefetch next chunk (30 KB contiguous) toward L1/L2 while computing.
        if (c < 2) {
            const float* nxt = S + (size_t)(b * RR + (c + 1) * JC) * DD;
            for (int f = tid * 32; f < JC * DD; f += 256 * 32)  // 128B/line
                __builtin_prefetch(nxt + f, 0, 1);
        }

        // This wave: fixed i, lanes sweep j within the chunk (4 strides of 32)
#pragma unroll
        for (int t = 0; t < 4; ++t) {
            const int jl = lane + 32 * t;
            if (jl < JC) {
                const float* sp = sS + jl * PADW;
                f32x2 acc0; acc0.x = 0.f; acc0.y = 0.f;
                f32x2 acc1; acc1.x = 0.f; acc1.y = 0.f;
#pragma unroll
                for (int q = 0; q < 16; ++q) {
                    V4u s; s.v = *((const v4f*)(sp + 4 * q));
                    f32x2 t0 = pk_add(s.h[0], r2[2 * q]);
                    f32x2 t1 = pk_add(s.h[1], r2[2 * q + 1]);
                    f32x2 h0; h0.x = fmaxf(t0.x, 0.f); h0.y = fmaxf(t0.y, 0.f);
                    f32x2 h1; h1.x = fmaxf(t1.x, 0.f); h1.y = fmaxf(t1.y, 0.f);
                    pk_fma(acc0, h0, w2[2 * q]);
                    pk_fma(acc1, h1, w2[2 * q + 1]);
                }
                const float acc = bc + ((acc0.x + acc0.y) + (acc1.x + acc1.y));
                orow[jbase + jl] = fmaxf(acc, 0.f);
            }
        }
    }
}

// ---------------------------------------------------------------------------
extern "C" void kernel_launch(void* const* d_in, const int* in_sizes, int n_in,
                              void* d_out, int out_size, void* d_ws, size_t ws_size,
                              hipStream_t stream)
{
    (void)in_sizes; (void)n_in; (void)out_size; (void)ws_size;

    const float* x    = (const float*)d_in[0];  // (16, 360, 64)
    const float* Wout = (const float*)d_in[1];  // (128, 64)
    const float* bout = (const float*)d_in[2];  // (64)
    const float* Wcat = (const float*)d_in[3];  // (64, 1)
    const float* bcat = (const float*)d_in[4];  // (1)

    float* Sws = (float*)d_ws;                  // 5760*64 floats
    float* Rws = Sws + (size_t)NN * DD;         // 5760*64 floats (~2.95 MB total)

    // Stage 1: 5760 rows / (4 waves * 16 rows) = 90 blocks of 128 threads
    stage1_gemm_wmma<<<NN / 64, 128, 0, stream>>>(x, Wout, bout, Sws, Rws);

    // Stage 2: one workgroup per (8 receivers, batch) -> 45 x 16 = 720 WGs
    dim3 g2(RR / IB, BB);
    stage2_pair<<<g2, 256, 0, stream>>>(Sws, Rws, Wcat, bcat, (float*)d_out);
}